// ConvGraph_SC_61521111548350
// MI455X (gfx1250) — compile-verified
//
#include <hip/hip_runtime.h>
#include <hip/hip_bf16.h>
#include <math.h>

// Problem constants (from reference): B=64, N=32, C=512, H=W=7
#define BATCH 64
#define NNODE 32
#define CDIM  512
#define HW    49

typedef __attribute__((ext_vector_type(2))) float v2f;
typedef __attribute__((ext_vector_type(8))) float v8f;

// -----------------------------------------------------------------------------
// Kernel 1: global average pool  input[B,N,C,7,7] -> state[B,N,C]
// Memory-bound: 205 MB single-use stream => non-temporal loads (don't evict the
// L2-resident weights/workspace), coalesced per-wave shuffle reduce.
// -----------------------------------------------------------------------------
__global__ __launch_bounds__(256) void pool_kernel(const float* __restrict__ in,
                                                   float* __restrict__ state) {
    const int bn   = blockIdx.x;              // 0 .. B*N-1
    const int lane = threadIdx.x & 31;
    const int wave = threadIdx.x >> 5;

    const float* base = in + (size_t)bn * CDIM * HW;
    float* outp = state + (size_t)bn * CDIM;

    for (int c = wave; c < CDIM; c += 8) {
        const float* p = base + c * HW;
        float v = __builtin_nontemporal_load(p + lane);
        if (lane < HW - 32) v += __builtin_nontemporal_load(p + 32 + lane);
        #pragma unroll
        for (int off = 16; off; off >>= 1) v += __shfl_xor(v, off);
        if (lane == 0) outp[c] = v * (1.0f / (float)HW);
    }
}

// -----------------------------------------------------------------------------
// Kernel 2: mat{1,2}[b] = state[b] @ W{1,2}^T + b{1,2}   via WMMA f32 16x16x4.
// Grid: 64 batches x 2 matrices x 4 d-groups = 512 blocks of 8 waves
// (~8 waves/SIMD so ds/global load latency is hidden by wave interleaving).
// Each wave owns one 16-wide d-tile: 2 M-tiles, K=512 in steps of 4.
// -----------------------------------------------------------------------------
__global__ __launch_bounds__(256) void gemm_kernel(const float* __restrict__ state,
                                                   const float* __restrict__ W1,
                                                   const float* __restrict__ b1,
                                                   const float* __restrict__ W2,
                                                   const float* __restrict__ b2,
                                                   float* __restrict__ mats) {
    extern __shared__ float sS[];             // 32*512 floats = 64 KB

    const int blk  = blockIdx.x;
    const int b    = blk >> 3;
    const int msel = (blk >> 2) & 1;          // which matrix (W1/W2)
    const int g    = blk & 3;                 // d-group (8 d-tiles each)
    const int tid  = threadIdx.x;
    const int lane = tid & 31;
    const int wave = tid >> 5;
    const int hi   = lane >> 4;               // lanes 16-31 hold K+2 (A/B layout)
    const int lm   = lane & 15;

    const float* Sg = state + (size_t)b * NNODE * CDIM;
    for (int i = tid; i < NNODE * CDIM; i += 256) sS[i] = Sg[i];
    __syncthreads();

    const float* W    = msel ? W2 : W1;
    const float* bias = msel ? b2 : b1;
    const int dbase   = (g * 8 + wave) * 16;  // this wave's 16 output columns

    // C init with bias broadcast: C[m,n] = bias[dbase+n], n = lane%16
    const float bb = bias[dbase + lm];
    v8f c0 = {bb, bb, bb, bb, bb, bb, bb, bb};
    v8f c1 = c0;

    const float* wr = W + (size_t)(dbase + lm) * CDIM;   // lane's output row of W

    for (int k0 = 0; k0 < CDIM; k0 += 4) {
        const int k = k0 + 2 * hi;
        v2f a0 = { sS[lm * CDIM + k],        sS[lm * CDIM + k + 1] };
        v2f a1 = { sS[(lm + 16) * CDIM + k], sS[(lm + 16) * CDIM + k + 1] };
        v2f wv = { wr[k], wr[k + 1] };
        c0 = __builtin_amdgcn_wmma_f32_16x16x4_f32(false, a0, false, wv, (short)0, c0, false, false);
        c1 = __builtin_amdgcn_wmma_f32_16x16x4_f32(false, a1, false, wv, (short)0, c1, false, false);
    }

    // Store D tiles: D layout vgpr v -> row v (lanes 0-15) / v+8 (lanes 16-31)
    float* outp = mats + ((size_t)msel * BATCH + b) * NNODE * CDIM;
    #pragma unroll
    for (int v = 0; v < 8; ++v) {
        const int r = v + 8 * hi;
        outp[(size_t)r * CDIM + dbase + lm]        = c0[v];
        outp[(size_t)(r + 16) * CDIM + dbase + lm] = c1[v];
    }
}

// -----------------------------------------------------------------------------
// Kernel 3: per-batch  adj = mat1 @ mat2^T  (WMMA, K=512 split over 8 waves),
// global mean/std (ddof=1), row softmax, and the collapsed output
//   out[c] = (1/32) * sum_m (1 + colsum(softadj)[m]) * state[m,c]
// -----------------------------------------------------------------------------
__global__ __launch_bounds__(256) void fuse_kernel(const float* __restrict__ state,
                                                   const float* __restrict__ mats,
                                                   float* __restrict__ out) {
    extern __shared__ float smem[];
    float* sM1  = smem;                        // 32*512 = 16384
    float* sM2  = sM1 + NNODE * CDIM;          // 16384
    float* sAdj = sM2 + NNODE * CDIM;          // 1024
    float* sRed = sAdj + NNODE * NNODE;        // 64

    const int b    = blockIdx.x;
    const int tid  = threadIdx.x;
    const int lane = tid & 31;
    const int wave = tid >> 5;
    const int hi   = lane >> 4;
    const int lm   = lane & 15;

    const float* M1g = mats + (size_t)b * NNODE * CDIM;
    const float* M2g = mats + ((size_t)BATCH + b) * NNODE * CDIM;
    for (int i = tid; i < NNODE * CDIM; i += 256) { sM1[i] = M1g[i]; sM2[i] = M2g[i]; }
    for (int i = tid; i < NNODE * NNODE; i += 256) sAdj[i] = 0.0f;
    __syncthreads();

    // adj partials: each wave handles 16 of the 128 k-steps (k range: wave*64..+63)
    v8f a00 = {}, a01 = {}, a10 = {}, a11 = {};
    const int kbeg = wave * 64;
    for (int k0 = kbeg; k0 < kbeg + 64; k0 += 4) {
        const int k = k0 + 2 * hi;
        v2f aa0 = { sM1[lm * CDIM + k],        sM1[lm * CDIM + k + 1] };
        v2f aa1 = { sM1[(lm + 16) * CDIM + k], sM1[(lm + 16) * CDIM + k + 1] };
        v2f bb0 = { sM2[lm * CDIM + k],        sM2[lm * CDIM + k + 1] };   // B[k][m]=M2[m][k]
        v2f bb1 = { sM2[(lm + 16) * CDIM + k], sM2[(lm + 16) * CDIM + k + 1] };
        a00 = __builtin_amdgcn_wmma_f32_16x16x4_f32(false, aa0, false, bb0, (short)0, a00, false, false);
        a01 = __builtin_amdgcn_wmma_f32_16x16x4_f32(false, aa0, false, bb1, (short)0, a01, false, false);
        a10 = __builtin_amdgcn_wmma_f32_16x16x4_f32(false, aa1, false, bb0, (short)0, a10, false, false);
        a11 = __builtin_amdgcn_wmma_f32_16x16x4_f32(false, aa1, false, bb1, (short)0, a11, false, false);
    }

    // reduce per-wave partials (ds_add_f32)
    #pragma unroll
    for (int v = 0; v < 8; ++v) {
        const int r = v + 8 * hi;                 // adjacency row n
        atomicAdd(&sAdj[r * 32 + lm],             a00[v]);
        atomicAdd(&sAdj[r * 32 + 16 + lm],        a01[v]);
        atomicAdd(&sAdj[(r + 16) * 32 + lm],      a10[v]);
        atomicAdd(&sAdj[(r + 16) * 32 + 16 + lm], a11[v]);
    }
    __syncthreads();

    // global mean / std (ddof=1) over the 1024 adj elements
    float s = 0.0f, ss = 0.0f;
    for (int i = tid; i < NNODE * NNODE; i += 256) {
        const float x = sAdj[i];
        s += x; ss += x * x;
    }
    #pragma unroll
    for (int off = 16; off; off >>= 1) { s += __shfl_xor(s, off); ss += __shfl_xor(ss, off); }
    if (lane == 0) { sRed[wave] = s; sRed[8 + wave] = ss; }
    __syncthreads();
    if (tid == 0) {
        float S1 = 0.0f, S2 = 0.0f;
        for (int w = 0; w < 8; ++w) { S1 += sRed[w]; S2 += sRed[8 + w]; }
        const float mean = S1 / 1024.0f;
        const float var  = (S2 - 1024.0f * mean * mean) / 1023.0f;   // unbiased
        sRed[16] = mean;
        sRed[17] = 1.0f / sqrtf(var);
    }
    __syncthreads();
    const float mean = sRed[16], inv_std = sRed[17];

    // row softmax (axis=2): wave handles 4 rows, lane = column
    for (int r = wave; r < NNODE; r += 8) {
        float z = (sAdj[r * 32 + lane] - mean) * inv_std;
        float mx = z;
        #pragma unroll
        for (int off = 16; off; off >>= 1) mx = fmaxf(mx, __shfl_xor(mx, off));
        const float e = __expf(z - mx);
        float se = e;
        #pragma unroll
        for (int off = 16; off; off >>= 1) se += __shfl_xor(se, off);
        sAdj[r * 32 + lane] = e / se;
    }
    __syncthreads();

    // column weights: w[m] = 1 + sum_n softadj[n][m]
    if (wave == 0) {
        float acc = 1.0f;
        #pragma unroll
        for (int n = 0; n < NNODE; ++n) acc += sAdj[n * 32 + lane];
        sRed[32 + lane] = acc;
    }
    __syncthreads();

    // out[b][c] = (1/32) * sum_m w[m] * state[b][m][c]   (state from L2)
    const float* Sg = state + (size_t)b * NNODE * CDIM;
    for (int c = tid; c < CDIM; c += 256) {
        float acc = 0.0f;
        #pragma unroll
        for (int mm = 0; mm < NNODE; ++mm) acc += sRed[32 + mm] * Sg[mm * CDIM + c];
        out[(size_t)b * CDIM + c] = acc * (1.0f / (float)NNODE);
    }
}

// -----------------------------------------------------------------------------
extern "C" void kernel_launch(void* const* d_in, const int* in_sizes, int n_in,
                              void* d_out, int out_size, void* d_ws, size_t ws_size,
                              hipStream_t stream) {
    const float* x   = (const float*)d_in[0];  // [B,N,C,7,7]
    const float* W1  = (const float*)d_in[1];  // [C,C]
    const float* b1v = (const float*)d_in[2];  // [C]
    const float* W2  = (const float*)d_in[3];  // [C,C]
    const float* b2v = (const float*)d_in[4];  // [C]
    float* out = (float*)d_out;                // [B,C]

    float* state = (float*)d_ws;                              // 4 MB
    float* mats  = state + (size_t)BATCH * NNODE * CDIM;      // 8 MB ([2][B][32][512])

    pool_kernel<<<BATCH * NNODE, 256, 0, stream>>>(x, state);

    gemm_kernel<<<BATCH * 8, 256, (size_t)NNODE * CDIM * sizeof(float), stream>>>(
        state, W1, b1v, W2, b2v, mats);

    const size_t smem3 =
        (size_t)(2 * NNODE * CDIM + NNODE * NNODE + 64) * sizeof(float);
    fuse_kernel<<<BATCH, 256, smem3, stream>>>(state, mats, out);

    (void)in_sizes; (void)n_in; (void)out_size; (void)ws_size;
}